// PairDist_40364102647835
// MI455X (gfx1250) — compile-verified
//
#include <hip/hip_runtime.h>

// CDNA5 / gfx1250, wave32. f32 WMMA path: V_WMMA_F32_16X16X4_F32.
typedef float v2f __attribute__((ext_vector_type(2)));
typedef float v8f __attribute__((ext_vector_type(8)));

#define D_DIM 128
#define BM 128
#define BN 128
#define LDS_STRIDE 132   // 128 + 4 floats pad: breaks the stride-128 bank aliasing (64 banks)
#define EPS 1e-6f

// ---------------- row squared-norms: sq[i] = sum_k x[i][k]^2 ----------------
__global__ __launch_bounds__(256) void pd_rowsq(const float* __restrict__ x,
                                                float* __restrict__ sq, int n) {
    int i = blockIdx.x * blockDim.x + threadIdx.x;
    if (i >= n) return;
    const float4* xr = (const float4*)(x + (size_t)i * D_DIM);
    float s = 0.0f;
#pragma unroll
    for (int q = 0; q < D_DIM / 4; ++q) {
        float4 v = xr[q];
        s += v.x * v.x + v.y * v.y + v.z * v.z + v.w * v.w;
    }
    sq[i] = s;
}

// ---------------- main: 128x128 tile of sqrt(max((sqi+sqj-2*x.xT)/D,0)+eps) ----------------
__global__ __launch_bounds__(256) void pd_main(const float* __restrict__ x,
                                               const float* __restrict__ sq,
                                               float* __restrict__ out, int n) {
    __shared__ float As[BM * LDS_STRIDE];   // rows [blockIdx.x*128, +128), all of K
    __shared__ float Bs[BN * LDS_STRIDE];   // rows [blockIdx.y*128, +128), all of K

    const int tid  = threadIdx.x;
    const int rowM = blockIdx.x * BM;
    const int rowN = blockIdx.y * BN;

    // Cooperative stage: 128 rows x 128 f32 per array = 4096 float4 each; 16 per thread.
    for (int idx = tid; idx < (BM * D_DIM) / 4; idx += 256) {
        int r  = idx >> 5;      // 32 float4 per row
        int kq = idx & 31;
        float4 va = *(const float4*)(x + (size_t)(rowM + r) * D_DIM + kq * 4);
        float4 vb = *(const float4*)(x + (size_t)(rowN + r) * D_DIM + kq * 4);
        *(float4*)(&As[r * LDS_STRIDE + kq * 4]) = va;
        *(float4*)(&Bs[r * LDS_STRIDE + kq * 4]) = vb;
    }
    __syncthreads();

    const int lane = tid & 31;
    const int wave = tid >> 5;          // 8 waves
    const int wm   = (wave & 3) * 32;   // wave row base: 4 waves x 32 rows
    const int wn   = (wave >> 2) * 64;  // wave col base: 2 waves x 64 cols

    // f32 WMMA 16x16x4 per-lane indexing (ISA 7.12.2):
    //   A 16x4: lane L -> M = L%16, holds K = k0+{0,1} (lanes<16) or k0+{2,3} (lanes>=16)
    //   B 4x16: lane L -> N = L%16, same K split. B = x^T, so both load row-major x strips.
    const int lrow  = lane & 15;
    const int khalf = (lane >> 4) << 1;     // 0 or 2

    v8f acc[2][4];
#pragma unroll
    for (int i = 0; i < 2; ++i)
#pragma unroll
        for (int j = 0; j < 4; ++j)
            acc[i][j] = (v8f){0.f, 0.f, 0.f, 0.f, 0.f, 0.f, 0.f, 0.f};

#pragma unroll 4
    for (int k = 0; k < D_DIM; k += 4) {
        const int kk = k + khalf;
        v2f a0 = *(const v2f*)(&As[(wm +  0 + lrow) * LDS_STRIDE + kk]);
        v2f a1 = *(const v2f*)(&As[(wm + 16 + lrow) * LDS_STRIDE + kk]);
        v2f b0 = *(const v2f*)(&Bs[(wn +  0 + lrow) * LDS_STRIDE + kk]);
        v2f b1 = *(const v2f*)(&Bs[(wn + 16 + lrow) * LDS_STRIDE + kk]);
        v2f b2 = *(const v2f*)(&Bs[(wn + 32 + lrow) * LDS_STRIDE + kk]);
        v2f b3 = *(const v2f*)(&Bs[(wn + 48 + lrow) * LDS_STRIDE + kk]);

        acc[0][0] = __builtin_amdgcn_wmma_f32_16x16x4_f32(false, a0, false, b0, (short)0, acc[0][0], false, false);
        acc[0][1] = __builtin_amdgcn_wmma_f32_16x16x4_f32(false, a0, false, b1, (short)0, acc[0][1], false, false);
        acc[0][2] = __builtin_amdgcn_wmma_f32_16x16x4_f32(false, a0, false, b2, (short)0, acc[0][2], false, false);
        acc[0][3] = __builtin_amdgcn_wmma_f32_16x16x4_f32(false, a0, false, b3, (short)0, acc[0][3], false, false);
        acc[1][0] = __builtin_amdgcn_wmma_f32_16x16x4_f32(false, a1, false, b0, (short)0, acc[1][0], false, false);
        acc[1][1] = __builtin_amdgcn_wmma_f32_16x16x4_f32(false, a1, false, b1, (short)0, acc[1][1], false, false);
        acc[1][2] = __builtin_amdgcn_wmma_f32_16x16x4_f32(false, a1, false, b2, (short)0, acc[1][2], false, false);
        acc[1][3] = __builtin_amdgcn_wmma_f32_16x16x4_f32(false, a1, false, b3, (short)0, acc[1][3], false, false);
    }

    // Epilogue. C/D 16x16 f32 layout: VGPR v, lane L -> M = v + 8*(L>=16), N = L%16.
    const float inv_d = 1.0f / (float)D_DIM;
    const int   mtop  = (lane >> 4) << 3;   // 0 or 8
#pragma unroll
    for (int i = 0; i < 2; ++i) {
        const int ig_base = rowM + wm + i * 16 + mtop;
#pragma unroll
        for (int j = 0; j < 4; ++j) {
            const int jg = rowN + wn + j * 16 + lrow;
            const float sqj = sq[jg];
#pragma unroll
            for (int v = 0; v < 8; ++v) {
                const int ig = ig_base + v;
                float g = acc[i][j][v];
                float q = (sq[ig] + sqj - 2.0f * g) * inv_d;
                q = fmaxf(q, 0.0f) + EPS;
                out[(size_t)ig * (size_t)n + (size_t)jg] = sqrtf(q);
            }
        }
    }
}

extern "C" void kernel_launch(void* const* d_in, const int* in_sizes, int n_in,
                              void* d_out, int out_size, void* d_ws, size_t ws_size,
                              hipStream_t stream) {
    const float* x = (const float*)d_in[0];
    const int n = in_sizes[0] / D_DIM;      // 8192
    float* sq  = (float*)d_ws;              // n floats of scratch
    float* out = (float*)d_out;

    pd_rowsq<<<dim3((n + 255) / 256), dim3(256), 0, stream>>>(x, sq, n);

    dim3 grid(n / BM, n / BN);
    pd_main<<<grid, dim3(256), 0, stream>>>(x, sq, out, n);
}